// GNNLayer_71854802862196
// MI455X (gfx1250) — compile-verified
//
#include <hip/hip_runtime.h>

typedef __attribute__((ext_vector_type(2))) float v2f;
typedef __attribute__((ext_vector_type(8))) float v8f;

#define DH 128
#define NTILE 8 // 128 / 16

// ---------------- degree kernels ----------------
__global__ void k_deg_init(float* __restrict__ deg, int N) {
  int i = blockIdx.x * blockDim.x + threadIdx.x;
  if (i < N) deg[i] = 1.0f; // self-loop contributes 1
}

__global__ void k_deg_edges(const long long* __restrict__ col,
                            float* __restrict__ deg, int E) {
  int e = blockIdx.x * blockDim.x + threadIdx.x;
  if (e < E) atomicAdd(&deg[(int)col[e]], 1.0f);
}

// ---- fused MLP1 over 32 rows/wave: hs = dinv * (relu(x@W1+b1) @ Wg); acc = hs
__global__ void __launch_bounds__(32) k_mlp1(
    const float* __restrict__ x, const float* __restrict__ W1,
    const float* __restrict__ b1, const float* __restrict__ Wg,
    const float* __restrict__ deg,
    float* __restrict__ hs, float* __restrict__ acc, int N) {
  __shared__ float lh[32 * DH];   // H1 staging, 16 KB
  __shared__ float sdinv[32];

  const int lane = threadIdx.x;        // 0..31
  const int r    = lane & 15;
  const int koff = (lane >> 4) * 2;    // 0 for lanes 0-15, 2 for lanes 16-31
  const int row0 = blockIdx.x * 32;

  // ---- load two A tiles (rows row0..+15 and row0+16..+31), f32 16x4 A layout
  int rlo = row0 + r;      if (rlo >= N) rlo = N - 1;
  int rhi = row0 + 16 + r; if (rhi >= N) rhi = N - 1;
  const float* xlo = x + (size_t)rlo * DH;
  const float* xhi = x + (size_t)rhi * DH;
  v2f alo[32], ahi[32];
#pragma unroll
  for (int g = 0; g < 32; ++g) {
    alo[g] = *(const v2f*)(xlo + 4 * g + koff);
    ahi[g] = *(const v2f*)(xhi + 4 * g + koff);
  }

  {
    int rr = row0 + lane; if (rr >= N) rr = N - 1;
    sdinv[lane] = rsqrtf(deg[rr]);
  }

  // ---- GEMM1: H1 = relu(x @ W1 + b1) -> LDS (row-major 32x128) ----
#pragma unroll 1
  for (int nt = 0; nt < NTILE; ++nt) {
    const int ncol = nt * 16 + r;
    v8f c0 = {}, c1 = {};
#pragma unroll
    for (int g = 0; g < 32; ++g) {
      v2f b;
      b.x = W1[(4 * g + koff) * DH + ncol];
      b.y = W1[(4 * g + koff + 1) * DH + ncol];
      c0 = __builtin_amdgcn_wmma_f32_16x16x4_f32(false, alo[g], false, b,
                                                 (short)0, c0, false, false);
      c1 = __builtin_amdgcn_wmma_f32_16x16x4_f32(false, ahi[g], false, b,
                                                 (short)0, c1, false, false);
    }
    const float bias  = b1[ncol];
    const int   crow0 = (lane >> 4) * 8;
#pragma unroll
    for (int v = 0; v < 8; ++v) {
      float v0 = c0[v] + bias;
      float v1 = c1[v] + bias;
      lh[(crow0 + v) * DH + ncol]      = v0 > 0.f ? v0 : 0.f;
      lh[(16 + crow0 + v) * DH + ncol] = v1 > 0.f ? v1 : 0.f;
    }
  }
  __syncthreads();

  // ---- reload H1 from LDS in A layout ----
  v2f blo[32], bhi[32];
#pragma unroll
  for (int g = 0; g < 32; ++g) {
    blo[g] = *(const v2f*)(&lh[r * DH + 4 * g + koff]);
    bhi[g] = *(const v2f*)(&lh[(16 + r) * DH + 4 * g + koff]);
  }

  // ---- GEMM2: HS = (H1 @ Wg) * dinv[row]; also init acc with self term ----
#pragma unroll 1
  for (int nt = 0; nt < NTILE; ++nt) {
    const int ncol = nt * 16 + r;
    v8f c0 = {}, c1 = {};
#pragma unroll
    for (int g = 0; g < 32; ++g) {
      v2f b;
      b.x = Wg[(4 * g + koff) * DH + ncol];
      b.y = Wg[(4 * g + koff + 1) * DH + ncol];
      c0 = __builtin_amdgcn_wmma_f32_16x16x4_f32(false, blo[g], false, b,
                                                 (short)0, c0, false, false);
      c1 = __builtin_amdgcn_wmma_f32_16x16x4_f32(false, bhi[g], false, b,
                                                 (short)0, c1, false, false);
    }
    const int crow0 = (lane >> 4) * 8;
#pragma unroll
    for (int v = 0; v < 8; ++v) {
      const int rr0 = crow0 + v;        // 0..15
      const int rr1 = 16 + crow0 + v;   // 16..31
      const int g0  = row0 + rr0;
      const int g1  = row0 + rr1;
      if (g0 < N) {
        float val = c0[v] * sdinv[rr0];
        hs[(size_t)g0 * DH + ncol]  = val;
        acc[(size_t)g0 * DH + ncol] = val; // self-loop contribution
      }
      if (g1 < N) {
        float val = c1[v] * sdinv[rr1];
        hs[(size_t)g1 * DH + ncol]  = val;
        acc[(size_t)g1 * DH + ncol] = val;
      }
    }
  }
}

// ---------------- edge scatter: acc[col] += hs[row], one wave per edge -----
__global__ void __launch_bounds__(256) k_scatter(
    const long long* __restrict__ row, const long long* __restrict__ col,
    const float* __restrict__ hs, float* __restrict__ acc, int E) {
  const int wid  = (int)((blockIdx.x * (size_t)blockDim.x + threadIdx.x) >> 5);
  const int lane = threadIdx.x & 31;
  if (wid >= E) return;
  const int s = (int)row[wid];
  const int t = (int)col[wid];
  const float4 v = *(const float4*)(hs + (size_t)s * DH + lane * 4);
  float* dst = acc + (size_t)t * DH + lane * 4;
  atomicAdd(dst + 0, v.x);
  atomicAdd(dst + 1, v.y);
  atomicAdd(dst + 2, v.z);
  atomicAdd(dst + 3, v.w);
}

// ------- MLP2 over 32 rows/wave: out = relu(dinv*acc + bg) @ W2 + b2 -------
__global__ void __launch_bounds__(32) k_mlp2(
    const float* __restrict__ acc, const float* __restrict__ deg,
    const float* __restrict__ bg, const float* __restrict__ W2,
    const float* __restrict__ b2, float* __restrict__ out, int N) {
  const int lane = threadIdx.x;
  const int r    = lane & 15;
  const int koff = (lane >> 4) * 2;
  const int row0 = blockIdx.x * 32;

  int rlo = row0 + r;      if (rlo >= N) rlo = N - 1;
  int rhi = row0 + 16 + r; if (rhi >= N) rhi = N - 1;
  const float dlo = rsqrtf(deg[rlo]);
  const float dhi = rsqrtf(deg[rhi]);
  const float* plo = acc + (size_t)rlo * DH;
  const float* phi = acc + (size_t)rhi * DH;

  v2f alo[32], ahi[32];
#pragma unroll
  for (int g = 0; g < 32; ++g) {
    v2f bgv = *(const v2f*)(bg + 4 * g + koff);
    v2f t0  = *(const v2f*)(plo + 4 * g + koff);
    v2f t1  = *(const v2f*)(phi + 4 * g + koff);
    float a0 = dlo * t0.x + bgv.x, a1 = dlo * t0.y + bgv.y;
    float b0 = dhi * t1.x + bgv.x, b1v = dhi * t1.y + bgv.y;
    alo[g].x = a0 > 0.f ? a0 : 0.f;
    alo[g].y = a1 > 0.f ? a1 : 0.f;
    ahi[g].x = b0 > 0.f ? b0 : 0.f;
    ahi[g].y = b1v > 0.f ? b1v : 0.f;
  }

#pragma unroll 1
  for (int nt = 0; nt < NTILE; ++nt) {
    const int ncol = nt * 16 + r;
    v8f c0 = {}, c1 = {};
#pragma unroll
    for (int g = 0; g < 32; ++g) {
      v2f b;
      b.x = W2[(4 * g + koff) * DH + ncol];
      b.y = W2[(4 * g + koff + 1) * DH + ncol];
      c0 = __builtin_amdgcn_wmma_f32_16x16x4_f32(false, alo[g], false, b,
                                                 (short)0, c0, false, false);
      c1 = __builtin_amdgcn_wmma_f32_16x16x4_f32(false, ahi[g], false, b,
                                                 (short)0, c1, false, false);
    }
    const float bias  = b2[ncol];
    const int   crow0 = (lane >> 4) * 8;
#pragma unroll
    for (int v = 0; v < 8; ++v) {
      const int g0 = row0 + crow0 + v;
      const int g1 = row0 + 16 + crow0 + v;
      if (g0 < N) out[(size_t)g0 * DH + ncol] = c0[v] + bias;
      if (g1 < N) out[(size_t)g1 * DH + ncol] = c1[v] + bias;
    }
  }
}

// ---------------------------------------------------------------------------
extern "C" void kernel_launch(void* const* d_in, const int* in_sizes, int n_in,
                              void* d_out, int out_size, void* d_ws, size_t ws_size,
                              hipStream_t stream) {
  const float*     x   = (const float*)d_in[0];
  const long long* ei  = (const long long*)d_in[1]; // int64 (2, E)
  const float*     W1  = (const float*)d_in[2];
  const float*     b1  = (const float*)d_in[3];
  const float*     Wg  = (const float*)d_in[4];
  const float*     bg  = (const float*)d_in[5];
  const float*     W2  = (const float*)d_in[6];
  const float*     b2  = (const float*)d_in[7];
  float*           out = (float*)d_out;

  const int N = in_sizes[0] / DH;
  const int E = in_sizes[1] / 2;
  const long long* rowp = ei;
  const long long* colp = ei + E;

  // workspace: deg[N] | hs[N*128] | acc[N*128]
  char*  ws  = (char*)d_ws;
  float* deg = (float*)ws;
  size_t off = (((size_t)N * sizeof(float)) + 255) & ~(size_t)255;
  float* hs  = (float*)(ws + off);
  off += (((size_t)N * DH * sizeof(float)) + 255) & ~(size_t)255;
  float* acc = (float*)(ws + off);

  const int ntiles = (N + 31) / 32;

  k_deg_init <<<(N + 255) / 256, 256, 0, stream>>>(deg, N);
  k_deg_edges<<<(E + 255) / 256, 256, 0, stream>>>(colp, deg, E);
  k_mlp1     <<<ntiles, 32, 0, stream>>>(x, W1, b1, Wg, deg, hs, acc, N);
  k_scatter  <<<(E + 7) / 8, 256, 0, stream>>>(rowp, colp, hs, acc, E);
  k_mlp2     <<<ntiles, 32, 0, stream>>>(acc, deg, bg, W2, b2, out, N);
}